// Top1Router_77335181132291
// MI455X (gfx1250) — compile-verified
//
#include <hip/hip_runtime.h>
#include <hip/hip_bf16.h>
#include <math.h>

typedef __attribute__((ext_vector_type(2))) float v2f;
typedef __attribute__((ext_vector_type(4))) float f4;
typedef __attribute__((ext_vector_type(8))) float v8f;

#define S_TOK 8192
#define N_EXP 64
#define CAP   256   // floor(2.0*8192/64)=256, even, >=4

// ---------------------------------------------------------------------------
// Kernel 1: per-token top1 expert + softmax probability of that expert.
// One wave32 handles 16 tokens. The softmax denominator sum_k exp(x[t][k])
// is computed on the matrix unit: V_WMMA_F32_16X16X4_F32 with B = ones
// yields D[m,n] = sum_k A[m,k] (row sums, replicated across N).
// A-matrix fp32 16x4 layout: lanes 0-15 supply K=0,1 (vgpr0,1) for row M=lane,
// lanes 16-31 supply K=2,3 for row M=lane-16 -> exactly how we load below.
// ---------------------------------------------------------------------------
__global__ __launch_bounds__(256) void k_route(const float* __restrict__ in,
                                               int* __restrict__ top1,
                                               float* __restrict__ prob) {
  const int lane    = threadIdx.x & 31;
  const int wave    = threadIdx.x >> 5;
  const int tokBase = blockIdx.x * 128 + wave * 16;
  const int hi      = lane >> 4;            // 0 -> K%4 in {0,1}; 1 -> {2,3}
  const float* p    = in + (size_t)(tokBase + (lane & 15)) * N_EXP;

  v8f acc = {};
  v2f b; b[0] = 1.0f; b[1] = 1.0f;          // ones matrix: layout-invariant
  float m = -1e30f; int mi = 0;

#pragma unroll
  for (int k0 = 0; k0 < N_EXP; k0 += 4) {
    const int kA = k0 + 2 * hi;
    float x0 = p[kA];
    float x1 = p[kA + 1];
    if (x0 > m) { m = x0; mi = kA; }
    if (x1 > m) { m = x1; mi = kA + 1; }
    v2f a; a[0] = expf(x0); a[1] = expf(x1);
    acc = __builtin_amdgcn_wmma_f32_16x16x4_f32(false, a, false, b,
                                                (short)0, acc, false, false);
  }

  // Combine argmax with partner lane (lane^16) covering the other K residues.
  // First-occurrence tie-break (matches jnp.argmax): smaller index wins.
  float om = __shfl_xor(m, 16, 32);
  int  omi = __shfl_xor(mi, 16, 32);
  if (om > m || (om == m && omi < mi)) { m = om; mi = omi; }

  // D layout: lane L, vgpr r holds D[M = r + 8*(L>=16), N = L%16].
  // Lane t (t<16) wants token t's sum: source lane s = (t<8)? t : t+8,
  // element r = t&7 = s&7.
  const int r = lane & 7;
  float pick = acc[0];
  pick = (r == 1) ? acc[1] : pick;
  pick = (r == 2) ? acc[2] : pick;
  pick = (r == 3) ? acc[3] : pick;
  pick = (r == 4) ? acc[4] : pick;
  pick = (r == 5) ? acc[5] : pick;
  pick = (r == 6) ? acc[6] : pick;
  pick = (r == 7) ? acc[7] : pick;
  const int src = (lane & 8) ? (lane + 8) : lane;
  float mySum = __shfl(pick, src, 32);

  if (lane < 16) {
    const int t = tokBase + lane;
    top1[t] = mi;
    prob[t] = expf(m) / mySum;   // == softmax(x)[argmax], max-shift cancels
  }
}

// ---------------------------------------------------------------------------
// Kernel 2: rank of each token within its expert (token-order inclusive
// cumsum - 1). One wave32 per expert; ballot + popcount prefix scan gives the
// exact sequential semantics. rank = -1 marks dropped (rank >= capacity).
// Every token matches exactly one expert, so rank[] is fully written.
// ---------------------------------------------------------------------------
__global__ __launch_bounds__(1024) void k_rank(const int* __restrict__ top1,
                                               int* __restrict__ rank) {
  const int lane   = threadIdx.x & 31;
  const int expert = blockIdx.x * 32 + (threadIdx.x >> 5);
  const unsigned lowmask = (1u << lane) - 1u;
  int base = 0;
  for (int c = 0; c < S_TOK; c += 32) {
    const int t = c + lane;
    const bool match = (top1[t] == expert);
    const unsigned bm = (unsigned)__ballot(match);
    if (match) {
      const int rk = base + __popc(bm & lowmask);
      rank[t] = (rk < CAP) ? rk : -1;
    }
    base += __popc(bm);
  }
}

// ---------------------------------------------------------------------------
// Kernel 3: zero-fill the 1 GiB output with 128-bit non-temporal stores.
// This is the bandwidth-bound bulk of the problem (~45 us at 23.3 TB/s).
// ---------------------------------------------------------------------------
__global__ __launch_bounds__(256) void k_fill(f4* __restrict__ out,
                                              long long n4, long long n) {
  long long i = (long long)blockIdx.x * blockDim.x + threadIdx.x;
  const long long stride = (long long)gridDim.x * blockDim.x;
  f4 z = {0.0f, 0.0f, 0.0f, 0.0f};
  for (long long j = i; j < n4; j += stride)
    __builtin_nontemporal_store(z, out + j);
  // generic tail (n % 4 elements)
  const long long tail = n4 * 4;
  if (i < n - tail)
    __builtin_nontemporal_store(0.0f, (float*)out + tail + i);
}

// ---------------------------------------------------------------------------
// Kernel 4: scatter the (at most) 8192 nonzeros: combine[t,e,rank] = prob,
// sec_mask[t,e,rank] = 1.0.
// ---------------------------------------------------------------------------
__global__ __launch_bounds__(256) void k_scatter(const int* __restrict__ top1,
                                                 const float* __restrict__ prob,
                                                 const int* __restrict__ rank,
                                                 float* __restrict__ comb,
                                                 float* __restrict__ sec) {
  const int t = blockIdx.x * blockDim.x + threadIdx.x;
  if (t >= S_TOK) return;
  const int rk = rank[t];
  if (rk < 0) return;
  const size_t off = ((size_t)t * N_EXP + top1[t]) * CAP + rk;
  comb[off] = prob[t];
  sec[off]  = 1.0f;
}

extern "C" void kernel_launch(void* const* d_in, const int* in_sizes, int n_in,
                              void* d_out, int out_size, void* d_ws, size_t ws_size,
                              hipStream_t stream) {
  (void)in_sizes; (void)n_in; (void)ws_size;
  const float* in = (const float*)d_in[0];

  int*   top1 = (int*)d_ws;
  float* prob = (float*)((char*)d_ws + (size_t)S_TOK * sizeof(int));
  int*   rank = (int*)((char*)d_ws + (size_t)S_TOK * (sizeof(int) + sizeof(float)));

  // K1: 64 blocks x 8 waves x 16 tokens = 8192 tokens
  k_route<<<64, 256, 0, stream>>>(in, top1, prob);

  // K2: 64 expert-waves across 2 blocks
  k_rank<<<2, 1024, 0, stream>>>(top1, rank);

  // K3: zero both outputs (combine_weights then sec_mask, concatenated)
  const long long n  = (long long)out_size;
  const long long n4 = n / 4;
  k_fill<<<2048, 256, 0, stream>>>((f4*)d_out, n4, n);

  // K4: scatter nonzeros
  float* comb = (float*)d_out;
  float* sec  = comb + (size_t)S_TOK * N_EXP * CAP;
  k_scatter<<<(S_TOK + 255) / 256, 256, 0, stream>>>(top1, prob, rank, comb, sec);
}